// retina_polar2_89017492177068
// MI455X (gfx1250) — compile-verified
//
#include <hip/hip_runtime.h>
#include <stdint.h>

// ---- problem constants (match reference) ----
namespace {
constexpr int Bc   = 64;
constexpr int Cc   = 3;
constexpr int HIc  = 512;
constexpr int WIc  = 512;
constexpr int HO   = 32;
constexpr int WO   = 64;
constexpr int UF   = 10;            // upsample factor (both dims)
constexpr int WUP  = WO * UF;       // 640
constexpr int PLANE = HIc * WIc;    // 262144 elements = 1 MB per channel plane
constexpr int TILE_FLOATS = UF * WUP * 2;            // 12800 floats = 51.2 KB
constexpr int TILE_CHUNKS = TILE_FLOATS * 4 / 16;    // 3200 x 16B
constexpr int BATCH_PER_BLK = 4;
constexpr int THREADS = BATCH_PER_BLK * WO;          // 256 threads = 8 waves (wave32)
}

__global__ __launch_bounds__(THREADS)
void retina_polar_glimpse_kernel(const float* __restrict__ x,
                                 const float* __restrict__ lt,
                                 const float* __restrict__ grid2d,
                                 float* __restrict__ out)
{
    // 10 rows of the upsampled polar grid, all 640 thetas, (gx,gy) pairs.
    __shared__ __align__(16) float tile[TILE_FLOATS];

    const int ro = blockIdx.x % HO;       // output radial row (0..31)
    const int bg = blockIdx.x / HO;       // batch group (0..15)
    const int sub = threadIdx.x >> 6;     // batch within group (0..3)
    const int to  = threadIdx.x & 63;     // output theta (0..63)
    const int b   = bg * BATCH_PER_BLK + sub;

    // ---- stage grid tile into LDS with CDNA5 async global->LDS copies ----
    {
        const uint64_t src = (uint64_t)(uintptr_t)grid2d +
                             (uint64_t)ro * (uint64_t)(UF * WUP * 2 * 4); // ro*51200B
        const uint32_t ldsbase = (uint32_t)(uintptr_t)(void*)tile;        // LDS byte offset
        for (int k = threadIdx.x; k < TILE_CHUNKS; k += THREADS) {
            const uint32_t ldsoff = ldsbase + (uint32_t)k * 16u;
            const uint32_t goff   = (uint32_t)k * 16u;
            asm volatile("global_load_async_to_lds_b128 %0, %1, %2"
                         :
                         : "v"(ldsoff), "v"(goff), "s"(src)
                         : "memory");
        }
        asm volatile("s_wait_asynccnt 0" ::: "memory");
    }
    __syncthreads();

    // ---- per-thread: one output pixel, all 3 channels ----
    const float lx = lt[2 * b + 0];
    const float ly = lt[2 * b + 1];
    // ix = ((gx + lx + 1)*512 - 1)*0.5 = gx*256 + (lx*256 + 255.5)
    const float ax = fmaf(lx, 256.0f, 255.5f);
    const float ay = fmaf(ly, 256.0f, 255.5f);

    const uint32_t base_b = (uint32_t)b * (uint32_t)(Cc * PLANE); // element offset of batch

    float a0 = 0.0f, a1 = 0.0f, a2 = 0.0f;
    const float2* __restrict__ t2 = reinterpret_cast<const float2*>(tile);

    for (int i = 0; i < UF; ++i) {
        const float2* __restrict__ row = t2 + i * WUP + to * UF;
#pragma unroll
        for (int j = 0; j < UF; ++j) {
            const float2 g = row[j];
            float ix = fmaf(g.x, 256.0f, ax);
            float iy = fmaf(g.y, 256.0f, ay);
            ix = fminf(fmaxf(ix, 0.0f), 511.0f);   // border padding (clamp)
            iy = fminf(fmaxf(iy, 0.0f), 511.0f);
            const float xf = floorf(ix);
            const float yf = floorf(iy);
            const float wx = ix - xf;
            const float wy = iy - yf;
            const int x0 = (int)xf;
            const int y0 = (int)yf;
            const int x1 = min(x0 + 1, WIc - 1);
            const int y1 = min(y0 + 1, HIc - 1);

            const uint32_t r0  = base_b + ((uint32_t)y0 << 9);
            const uint32_t r1  = base_b + ((uint32_t)y1 << 9);
            const uint32_t i00 = r0 + (uint32_t)x0;
            const uint32_t i01 = r0 + (uint32_t)x1;
            const uint32_t i10 = r1 + (uint32_t)x0;
            const uint32_t i11 = r1 + (uint32_t)x1;

            const float wx1 = 1.0f - wx;
            const float wy1 = 1.0f - wy;
            const float w00 = wx1 * wy1;
            const float w01 = wx  * wy1;
            const float w10 = wx1 * wy;
            const float w11 = wx  * wy;

            // channel 0 (ioffset 0), channel 1 (+1 MB), channel 2 (+2 MB):
            // same u32 voffsets, immediate ioffsets -> 1 address calc, 12 loads.
            {
                const float v00 = x[i00], v01 = x[i01], v10 = x[i10], v11 = x[i11];
                a0 = fmaf(v00, w00, fmaf(v01, w01, fmaf(v10, w10, fmaf(v11, w11, a0))));
            }
            {
                const float v00 = x[i00 + (uint32_t)PLANE], v01 = x[i01 + (uint32_t)PLANE];
                const float v10 = x[i10 + (uint32_t)PLANE], v11 = x[i11 + (uint32_t)PLANE];
                a1 = fmaf(v00, w00, fmaf(v01, w01, fmaf(v10, w10, fmaf(v11, w11, a1))));
            }
            {
                const float v00 = x[i00 + 2u * PLANE], v01 = x[i01 + 2u * PLANE];
                const float v10 = x[i10 + 2u * PLANE], v11 = x[i11 + 2u * PLANE];
                a2 = fmaf(v00, w00, fmaf(v01, w01, fmaf(v10, w10, fmaf(v11, w11, a2))));
            }
        }
    }

    // ---- write [B,C,HO,WO], channel stride HO*WO ----
    const float inv = 1.0f / (float)(UF * UF);
    float* __restrict__ o = out + (size_t)b * (Cc * HO * WO) + (size_t)ro * WO + to;
    o[0]              = a0 * inv;
    o[HO * WO]        = a1 * inv;
    o[2 * HO * WO]    = a2 * inv;
}

extern "C" void kernel_launch(void* const* d_in, const int* in_sizes, int n_in,
                              void* d_out, int out_size, void* d_ws, size_t ws_size,
                              hipStream_t stream) {
    (void)in_sizes; (void)n_in; (void)d_ws; (void)ws_size; (void)out_size;
    const float* x      = (const float*)d_in[0];   // [64,3,512,512]
    const float* lt     = (const float*)d_in[1];   // [64,2]
    const float* grid2d = (const float*)d_in[2];   // [320,640,2]
    float* out = (float*)d_out;                    // [64,3,32,64]

    dim3 grid((Bc / BATCH_PER_BLK) * HO);          // 512 blocks
    dim3 block(THREADS);                           // 256 threads = 8 wave32
    retina_polar_glimpse_kernel<<<grid, block, 0, stream>>>(x, lt, grid2d, out);
}